// NeuralODEFlow_15006615733266
// MI455X (gfx1250) — compile-verified
//
#include <hip/hip_runtime.h>

// NeuralODEFlow on gfx1250 (MI455X): fp32 WMMA (v_wmma_f32_16x16x4_f32) for the
// Jacobian-trace finite-difference pass, which dominates the FLOPs.
//
// DIM=64, HID=512, BATCH=1024, 9 Euler steps, H=1e-5.

typedef float v2f __attribute__((ext_vector_type(2)));
typedef float v8f __attribute__((ext_vector_type(8)));

#define DIM   64
#define HID   512
#define BATCH 1024
#define NSTEP 9
#define FDH   1e-5f

__device__ __forceinline__ v8f wmma_f32_k4(v2f a, v2f b, v8f c) {
  // 8 args: (neg_a, A, neg_b, B, c_mod, C, reuse_a, reuse_b)
  return __builtin_amdgcn_wmma_f32_16x16x4_f32(false, a, false, b, (short)0, c,
                                               false, false);
}

// ---------------------------------------------------------------------------
// init: z <- x ; log_det <- 0
// ---------------------------------------------------------------------------
__global__ void node_init(const float* __restrict__ x, float* __restrict__ z,
                          float* __restrict__ ld) {
  int i = blockIdx.x * 256 + threadIdx.x;
  if (i < BATCH * DIM) z[i] = x[i];
  else if (i < BATCH * DIM + BATCH) ld[i - BATCH * DIM] = 0.0f;
}

// ---------------------------------------------------------------------------
// base pass: dz = mlp(z)*dt ; z += dz.  8 rows per block, VALU (1.5% of FLOPs).
// ---------------------------------------------------------------------------
__global__ __launch_bounds__(256) void node_base(
    float* __restrict__ z, float* __restrict__ dz,
    const float* __restrict__ W1, const float* __restrict__ b1,
    const float* __restrict__ W2, const float* __restrict__ b2,
    const float* __restrict__ W3, const float* __restrict__ b3, float dt) {
  __shared__ float zs[8 * DIM];
  __shared__ float h1s[8 * HID];
  __shared__ float h2s[8 * HID];
  const int tid = threadIdx.x;
  const int row0 = blockIdx.x * 8;

  for (int i = tid; i < 8 * DIM; i += 256) zs[i] = z[row0 * DIM + i];
  __syncthreads();

  for (int idx = tid; idx < 8 * HID; idx += 256) {
    int r = idx >> 9, c = idx & (HID - 1);
    float s = b1[c];
    for (int i = 0; i < DIM; ++i) s += zs[r * DIM + i] * W1[i * HID + c];
    h1s[idx] = s > 0.0f ? s : 0.0f;
  }
  __syncthreads();

  for (int idx = tid; idx < 8 * HID; idx += 256) {
    int r = idx >> 9, c = idx & (HID - 1);
    float s = b2[c];
    for (int i = 0; i < HID; ++i) s += h1s[r * HID + i] * W2[i * HID + c];
    h2s[idx] = s > 0.0f ? s : 0.0f;
  }
  __syncthreads();

  for (int idx = tid; idx < 8 * DIM; idx += 256) {
    int r = idx >> 6, c = idx & (DIM - 1);
    float s = b3[c];
    for (int i = 0; i < HID; ++i) s += h2s[r * HID + i] * W3[i * DIM + c];
    float d = s * dt;
    dz[row0 * DIM + idx] = d;
    z[row0 * DIM + idx] += d;
  }
}

// ---------------------------------------------------------------------------
// trace pass: one block per batch element b. 256 threads = 8 waves.
//   Zp = z_b broadcast + H*I  (64x64)
//   H1 = relu(Zp @ W1 + b1)   (64x512, computed in 8 K-panels via WMMA)
//   H2 = relu(H1 @ W2 + b2)   (accumulated in registers, 64x64 per wave)
//   trace_b = (dt*(sum_j h2_j . W3[:,j] + sum b3) - sum dz_b) / H
//   log_det[b] -= trace_b
// ---------------------------------------------------------------------------
__global__ __launch_bounds__(256) void node_trace(
    const float* __restrict__ z, const float* __restrict__ dz,
    float* __restrict__ ld,
    const float* __restrict__ W1, const float* __restrict__ b1,
    const float* __restrict__ W2, const float* __restrict__ b2,
    const float* __restrict__ W3, const float* __restrict__ b3, float dt) {
  __shared__ float zp[64 * 65];   // perturbation tile, stride 65 (bank-safe)
  __shared__ float h1p[64 * 65];  // H1 K-panel, stride 65
  __shared__ float red[256];
  __shared__ float zb[DIM];
  __shared__ float dzb[DIM];

  const int b = blockIdx.x;
  const int tid = threadIdx.x;
  const int wave = tid >> 5;
  const int lane = tid & 31;
  const int half = lane >> 4;   // 0: lanes 0-15, 1: lanes 16-31
  const int lq = lane & 15;

  if (tid < DIM) {
    zb[tid] = z[b * DIM + tid];
    dzb[tid] = dz[b * DIM + tid];
  }
  __syncthreads();

  for (int idx = tid; idx < 64 * 64; idx += 256) {
    int j = idx >> 6, i = idx & 63;
    zp[j * 65 + i] = zb[i] + ((i == j) ? FDH : 0.0f);
  }
  __syncthreads();

  // H2 accumulators: wave owns hidden cols [wave*64, wave*64+64), all 64 rows.
  v8f acc[4][4];
#pragma unroll
  for (int mt = 0; mt < 4; ++mt)
#pragma unroll
    for (int nt = 0; nt < 4; ++nt) acc[mt][nt] = (v8f){0, 0, 0, 0, 0, 0, 0, 0};

  for (int kp = 0; kp < HID / 64; ++kp) {
    // --- layer-1 panel: h1p[0:64][0:64] = relu(Zp @ W1[:, kp*64 : +64] + b1)
#pragma unroll
    for (int ti = 0; ti < 2; ++ti) {
      int t = wave * 2 + ti;          // 16 tiles, 2 per wave
      int m0 = (t >> 2) * 16;
      int n0 = (t & 3) * 16;
      int cglob = kp * 64 + n0 + lq;  // W1 column for this lane
      v8f c = (v8f){0, 0, 0, 0, 0, 0, 0, 0};
#pragma unroll
      for (int ks = 0; ks < 16; ++ks) {
        int k = ks * 4 + 2 * half;
        v2f a, bb;
        a.x = zp[(m0 + lq) * 65 + k];
        a.y = zp[(m0 + lq) * 65 + k + 1];
        bb.x = W1[k * HID + cglob];
        bb.y = W1[(k + 1) * HID + cglob];
        c = wmma_f32_k4(a, bb, c);
      }
      float bias = b1[cglob];
#pragma unroll
      for (int r = 0; r < 8; ++r) {
        int m = m0 + r + 8 * half;
        float v = c[r] + bias;
        h1p[m * 65 + n0 + lq] = v > 0.0f ? v : 0.0f;
      }
    }
    __syncthreads();

    // --- layer-2 accumulate over this K-panel
#pragma unroll
    for (int ks = 0; ks < 16; ++ks) {
      int kl = ks * 4 + 2 * half;      // local K within panel
      int kg = kp * 64 + kl;           // global K
      v2f a[4], bb[4];
#pragma unroll
      for (int mt = 0; mt < 4; ++mt) {
        a[mt].x = h1p[(mt * 16 + lq) * 65 + kl];
        a[mt].y = h1p[(mt * 16 + lq) * 65 + kl + 1];
      }
#pragma unroll
      for (int nt = 0; nt < 4; ++nt) {
        int n = wave * 64 + nt * 16 + lq;
        bb[nt].x = W2[kg * HID + n];
        bb[nt].y = W2[(kg + 1) * HID + n];
      }
#pragma unroll
      for (int nt = 0; nt < 4; ++nt)
#pragma unroll
        for (int mt = 0; mt < 4; ++mt)
          acc[mt][nt] = wmma_f32_k4(a[mt], bb[nt], acc[mt][nt]);
    }
    __syncthreads();  // h1p reused next panel
  }

  // --- relu(+b2) and diagonal dot with W3: partial = sum h2[j][k]*W3[k][j]
  float p = 0.0f;
#pragma unroll
  for (int nt = 0; nt < 4; ++nt) {
    int col = wave * 64 + nt * 16 + lq;  // hidden index k
    float bias2 = b2[col];
#pragma unroll
    for (int mt = 0; mt < 4; ++mt) {
#pragma unroll
      for (int r = 0; r < 8; ++r) {
        int j = mt * 16 + r + 8 * half;  // output/perturbation index
        float h2v = acc[mt][nt][r] + bias2;
        h2v = h2v > 0.0f ? h2v : 0.0f;
        p += h2v * W3[col * DIM + j];
      }
    }
  }
  red[tid] = p;
  __syncthreads();
  for (int s = 128; s > 0; s >>= 1) {
    if (tid < s) red[tid] += red[tid + s];
    __syncthreads();
  }
  if (tid == 0) {
    float TD = red[0];
    float sb3 = 0.0f, sdz = 0.0f;
    for (int j = 0; j < DIM; ++j) {
      sb3 += b3[j];
      sdz += dzb[j];
    }
    float trace = (dt * (TD + sb3) - sdz) * (1.0f / FDH);
    ld[b] -= trace;
  }
}

// ---------------------------------------------------------------------------
// emit outputs: [z (1024x64), log_det (1024)]
// ---------------------------------------------------------------------------
__global__ void node_final(const float* __restrict__ z,
                           const float* __restrict__ ld,
                           float* __restrict__ out) {
  int i = blockIdx.x * 256 + threadIdx.x;
  if (i < BATCH * DIM) out[i] = z[i];
  else if (i < BATCH * DIM + BATCH) out[i] = ld[i - BATCH * DIM];
}

extern "C" void kernel_launch(void* const* d_in, const int* in_sizes, int n_in,
                              void* d_out, int out_size, void* d_ws,
                              size_t ws_size, hipStream_t stream) {
  const float* x  = (const float*)d_in[0];
  const float* W1 = (const float*)d_in[1];
  const float* b1 = (const float*)d_in[2];
  const float* W2 = (const float*)d_in[3];
  const float* b2 = (const float*)d_in[4];
  const float* W3 = (const float*)d_in[5];
  const float* b3 = (const float*)d_in[6];
  float* out = (float*)d_out;

  float* ws = (float*)d_ws;
  float* z  = ws;                    // 65536 floats
  float* dz = ws + BATCH * DIM;      // 65536 floats
  float* ld = ws + 2 * BATCH * DIM;  // 1024 floats

  const int ntot = BATCH * DIM + BATCH;  // 66560
  const int nblk = (ntot + 255) / 256;

  node_init<<<nblk, 256, 0, stream>>>(x, z, ld);

  const float dt = 1.0f / (float)NSTEP;  // linspace(0,1,10) -> uniform 1/9
  for (int s = 0; s < NSTEP; ++s) {
    node_base<<<BATCH / 8, 256, 0, stream>>>(z, dz, W1, b1, W2, b2, W3, b3, dt);
    node_trace<<<BATCH, 256, 0, stream>>>(z, dz, ld, W1, b1, W2, b2, W3, b3, dt);
  }
  node_final<<<nblk, 256, 0, stream>>>(z, ld, out);
}